// GraphSAGE_85160611545330
// MI455X (gfx1250) — compile-verified
//
#include <hip/hip_runtime.h>

typedef __attribute__((ext_vector_type(2))) float v2f;
typedef __attribute__((ext_vector_type(8))) float v8f;

#define N_NODES 100000
#define NFEATS  256
#define HIDDEN  128
#define NCLASS  64
#define BATCH   1024
#define K0      25
#define K1      10
#define NHIST   5
#define ROWS0   (BATCH + BATCH * K1)   // 11264 rows through layer 0
#define KDIM0   (2 * NFEATS)           // 512
#define KDIM1   (2 * HIDDEN)           // 256

// ---------------------------------------------------------------------------
// Stage A: Xcat0[row] = [ feats[self] (256) | mean_k feats[neigh[k]] (256) ]
// rows 0..1023   : self = nodes[row],      neigh = neighs0[row]
// rows 1024..end : self = neighs1_flat[i], neigh = neighs0_nb[i]
// 64 lanes per row, float4 loads; prefetch next gather row (feats table is
// 100MB < 192MB L2, so these gathers are latency-bound, not HBM-bound).
// ---------------------------------------------------------------------------
__global__ __launch_bounds__(256) void agg0_kernel(
    const float* __restrict__ feats,
    const int* __restrict__ nodes,
    const int* __restrict__ neighs0,
    const int* __restrict__ neighs1,
    const int* __restrict__ neighs0_nb,
    float* __restrict__ Xcat)
{
    const int row = blockIdx.x * 4 + (threadIdx.x >> 6);
    const int t   = threadIdx.x & 63;               // 64 float4 per 256-f row

    int self_idx;
    const int* __restrict__ neigh;
    if (row < BATCH) {
        self_idx = nodes[row];
        neigh    = neighs0 + row * K0;
    } else {
        const int i = row - BATCH;                   // i = b*K1 + k (b-major)
        self_idx = neighs1[i];
        neigh    = neighs0_nb + i * K0;
    }

    const float4* __restrict__ fp = (const float4*)feats; // 64 float4 per row
    const float4 self = fp[(size_t)self_idx * 64 + t];

    float ax = 0.f, ay = 0.f, az = 0.f, aw = 0.f;
    int idx = neigh[0];
    #pragma unroll 5
    for (int k = 0; k < K0; ++k) {
        int nxt = idx;
        if (k + 1 < K0) {
            nxt = neigh[k + 1];
            __builtin_prefetch(&fp[(size_t)nxt * 64 + t], 0, 3); // global_prefetch_b8
        }
        const float4 v = fp[(size_t)idx * 64 + t];
        ax += v.x; ay += v.y; az += v.z; aw += v.w;
        idx = nxt;
    }
    const float inv = 1.0f / (float)K0;
    float4 acc; acc.x = ax * inv; acc.y = ay * inv; acc.z = az * inv; acc.w = aw * inv;

    float4* __restrict__ xc = (float4*)(Xcat + (size_t)row * KDIM0);
    xc[t]      = self;   // first 256 floats: h_prev
    xc[64 + t] = acc;    // next 256 floats: neighbor mean
}

// ---------------------------------------------------------------------------
// fp32 WMMA GEMM, N-fused + LDS-staged B:
//   C(M x N) = relu(A(M x K) @ B(K x N) + bias)
// Block = 8 waves = 128 rows of A. Each wave owns a 16-row stripe and keeps
// N/16 accumulator tiles in VGPRs (A fragment reused across all column
// tiles).  B is staged one 64 x N K-slab at a time into LDS by the whole
// block (coalesced float4), so B global traffic is once-per-block.
// WMMA operand layout per CDNA5 ISA:
//   A (16x4 f32): lanes 0-15 row=lane hold K=k,k+1; lanes 16-31 K=k+2,k+3
//   B (4x16 f32): mirrored;  C/D: VGPR r -> M=r (+8 for lanes 16-31), N=lane&15
// Control flow is wave-uniform; EXEC is all-ones at every WMMA.
// ---------------------------------------------------------------------------
template <int K, int N>
__global__ __launch_bounds__(256) void wmma_gemm_fused(
    const float* __restrict__ A,
    const float* __restrict__ Bm,
    const float* __restrict__ bias,
    float* __restrict__ C)
{
    constexpr int NT    = N / 16;   // column tiles per wave (8 or 4)
    constexpr int KT    = 64;       // K-slab staged in LDS
    constexpr int WAVES = 8;
    __shared__ float Bs[KT * N];    // 32KB (N=128) / 16KB (N=64)

    const int wave = threadIdx.x >> 5;
    const int lane = threadIdx.x & 31;
    const int half = lane >> 4;     // 0: K=k..k+1, 1: K=k+2..k+3
    const int l15  = lane & 15;

    const int rowBase = blockIdx.x * (WAVES * 16) + wave * 16;
    const float* __restrict__ Arow = A + (size_t)(rowBase + l15) * K;

    v8f zero = {};
    v8f acc[NT];
    #pragma unroll
    for (int t = 0; t < NT; ++t) acc[t] = zero;

    for (int kt = 0; kt < K; kt += KT) {
        __syncthreads();            // protect Bs from previous slab's readers
        // Cooperative stage of B[kt..kt+KT) x N into LDS (coalesced float4).
        constexpr int TOT4 = KT * N / 4;   // 2048 or 1024; multiple of 256
        const float4* __restrict__ Bg = (const float4*)(Bm + (size_t)kt * N);
        float4* __restrict__ Bs4 = (float4*)Bs;
        #pragma unroll
        for (int i = threadIdx.x; i < TOT4; i += 256) Bs4[i] = Bg[i];
        __syncthreads();

        #pragma unroll 4
        for (int k = 0; k < KT; k += 4) {
            const int ka = k + 2 * half;
            v2f a; a.x = Arow[kt + ka]; a.y = Arow[kt + ka + 1];
            #pragma unroll
            for (int t = 0; t < NT; ++t) {
                const int c = t * 16 + l15;
                v2f b; b.x = Bs[ka * N + c]; b.y = Bs[(ka + 1) * N + c];
                acc[t] = __builtin_amdgcn_wmma_f32_16x16x4_f32(
                    /*neg_a=*/false, a, /*neg_b=*/false, b,
                    /*c_mod=*/(short)0, acc[t],
                    /*reuse_a=*/false, /*reuse_b=*/false);
            }
        }
    }

    // Fused bias + ReLU epilogue, one 16x16 tile at a time.
    #pragma unroll
    for (int t = 0; t < NT; ++t) {
        const int c  = t * 16 + l15;
        const float bv = bias[c];
        float* __restrict__ Crow = C + (size_t)(rowBase + 8 * half) * N + c;
        #pragma unroll
        for (int r = 0; r < 8; ++r) {
            float v = acc[t][r] + bv;
            Crow[(size_t)r * N] = v > 0.0f ? v : 0.0f;
        }
    }
}

// ---------------------------------------------------------------------------
// Stage C: second-layer input.
// agg[b] = ( 0.5 * sum_k (H_nb[b,k] - history[neighs1[b,k]])
//            + sum_j history[h_nodes[b,j]] ) / 15
// Xcat1[b] = [ h1[b] (128) | agg[b] (128) ]
// ---------------------------------------------------------------------------
__global__ __launch_bounds__(128) void agg1_kernel(
    const float* __restrict__ H,         // ROWS0 x 128 (0..1023 = h1, rest = neigh_original)
    const float* __restrict__ history,   // N_NODES x 128
    const int* __restrict__ neighs1,     // B*K1
    const int* __restrict__ h_nodes,     // B*NHIST
    float* __restrict__ Xcat1)           // B x 256
{
    const int b = blockIdx.x;
    const int t = threadIdx.x;           // 0..127

    float s = 0.f;
    #pragma unroll
    for (int k = 0; k < K1; ++k) {
        const int hn = neighs1[b * K1 + k];
        s += H[(size_t)(BATCH + b * K1 + k) * HIDDEN + t]
           - history[(size_t)hn * HIDDEN + t];
    }
    s *= (float)NHIST / (float)K1;       // * 0.5
    #pragma unroll
    for (int j = 0; j < NHIST; ++j) {
        const int hn = h_nodes[b * NHIST + j];
        s += history[(size_t)hn * HIDDEN + t];
    }
    s *= 1.0f / (float)(K1 + NHIST);     // / 15

    Xcat1[(size_t)b * KDIM1 + t]          = H[(size_t)b * HIDDEN + t];
    Xcat1[(size_t)b * KDIM1 + HIDDEN + t] = s;
}

// ---------------------------------------------------------------------------
extern "C" void kernel_launch(void* const* d_in, const int* in_sizes, int n_in,
                              void* d_out, int out_size, void* d_ws, size_t ws_size,
                              hipStream_t stream)
{
    const float* feats      = (const float*)d_in[0];
    const float* history    = (const float*)d_in[1];
    const float* W0         = (const float*)d_in[2];   // 512 x 128
    const float* b0         = (const float*)d_in[3];   // 128
    const float* W1         = (const float*)d_in[4];   // 256 x 64
    const float* b1         = (const float*)d_in[5];   // 64
    const int*   nodes      = (const int*)d_in[6];
    const int*   neighs0    = (const int*)d_in[7];
    const int*   neighs1    = (const int*)d_in[8];
    const int*   neighs0_nb = (const int*)d_in[9];
    const int*   h_nodes    = (const int*)d_in[10];
    float* out = (float*)d_out;

    // Workspace layout (~29.9 MB total)
    float* Xcat0 = (float*)d_ws;                         // ROWS0 * 512
    float* H     = Xcat0 + (size_t)ROWS0 * KDIM0;        // ROWS0 * 128
    float* Xcat1 = H     + (size_t)ROWS0 * HIDDEN;       // BATCH * 256
    (void)in_sizes; (void)n_in; (void)out_size; (void)ws_size;

    // Stage A: gather + mean -> Xcat0 (11264 x 512)
    agg0_kernel<<<dim3(ROWS0 / 4), 256, 0, stream>>>(
        feats, nodes, neighs0, neighs1, neighs0_nb, Xcat0);

    // Stage B: H = relu(Xcat0 @ W0 + b0)   (11264 x 128); 128 rows per block
    wmma_gemm_fused<KDIM0, HIDDEN>
        <<<dim3(ROWS0 / 128), 256, 0, stream>>>(Xcat0, W0, b0, H);

    // Stage C: delta/history aggregate -> Xcat1 (1024 x 256)
    agg1_kernel<<<dim3(BATCH), 128, 0, stream>>>(H, history, neighs1, h_nodes, Xcat1);

    // Stage D: out = relu(Xcat1 @ W1 + b1)  (1024 x 64); 128 rows per block
    wmma_gemm_fused<KDIM1, NCLASS>
        <<<dim3(BATCH / 128), 256, 0, stream>>>(Xcat1, W1, b1, out);
}